// CenterLoss_20323785245021
// MI455X (gfx1250) — compile-verified
//
#include <hip/hip_runtime.h>
#include <stdint.h>

#define NUM_CLASSES 100000
#define FEAT_DIM    256
#define BATCH       8192
#define ALPHA_C     1.0f
#define EPS_C       1e-6f

typedef __attribute__((ext_vector_type(2))) float v2f;
typedef __attribute__((ext_vector_type(8))) float v8f;

// ---------------------------------------------------------------------------
// Kernel 1: zero the per-class count array (d_ws) and the loss accumulator.
// ---------------------------------------------------------------------------
__global__ void cl_zero_kernel(float* __restrict__ counts, float* __restrict__ out_loss) {
    int i = blockIdx.x * blockDim.x + threadIdx.x;
    if (i < NUM_CLASSES) counts[i] = 0.0f;
    if (i == 0) out_loss[0] = 0.0f;
}

// ---------------------------------------------------------------------------
// Kernel 2: per-class appearance counts (float, like the reference segment_sum).
// ---------------------------------------------------------------------------
__global__ void cl_count_kernel(const int* __restrict__ target, float* __restrict__ counts) {
    int i = blockIdx.x * blockDim.x + threadIdx.x;
    if (i < BATCH) atomicAdd(&counts[target[i]], 1.0f);
}

// ---------------------------------------------------------------------------
// Kernel 3: stream centers (102.4 MB) -> new_centers via the CDNA5 async
// global<->LDS path (ASYNCcnt). Each lane moves 64 B per wait-window:
//   4x global_load_async_to_lds_b128 (16B-aligned src)
//   s_wait_asynccnt 0
//   16x global_store_async_from_lds_b32 (dst = d_out+1 is only 4B-aligned)
//   s_wait_asynccnt 0   (staging buffer reusable next iteration)
// LDS is laid out stage[tid][4] so each thread's 64 B are contiguous in LDS;
// the ISA applies the instruction offset to BOTH the LDS and the global
// address, so one LDS-base VGPR + offset: covers all 20 async ops.
// ---------------------------------------------------------------------------
__global__ void cl_copy_kernel(const float* __restrict__ src, float* __restrict__ dst, int n16) {
    __shared__ float4 stage[256][4];
    const uint32_t lds = (uint32_t)(uintptr_t)(void*)&stage[threadIdx.x][0];
    int i = blockIdx.x * blockDim.x + threadIdx.x;
    const int stride = gridDim.x * blockDim.x;
    for (; i < n16; i += stride) {
        const float* sp = src + (size_t)i * 16;   // 64B chunk, 16B aligned
        float*       dp = dst + (size_t)i * 16;   // 64B chunk, 4B aligned
        asm volatile("global_load_async_to_lds_b128 %0, %1, off"
                     :: "v"(lds), "v"(sp) : "memory");
        asm volatile("global_load_async_to_lds_b128 %0, %1, off offset:16"
                     :: "v"(lds), "v"(sp) : "memory");
        asm volatile("global_load_async_to_lds_b128 %0, %1, off offset:32"
                     :: "v"(lds), "v"(sp) : "memory");
        asm volatile("global_load_async_to_lds_b128 %0, %1, off offset:48"
                     :: "v"(lds), "v"(sp) : "memory");
        asm volatile("s_wait_asynccnt 0x0" ::: "memory");
        asm volatile("global_store_async_from_lds_b32 %0, %1, off"
                     :: "v"(dp), "v"(lds) : "memory");
        asm volatile("global_store_async_from_lds_b32 %0, %1, off offset:4"
                     :: "v"(dp), "v"(lds) : "memory");
        asm volatile("global_store_async_from_lds_b32 %0, %1, off offset:8"
                     :: "v"(dp), "v"(lds) : "memory");
        asm volatile("global_store_async_from_lds_b32 %0, %1, off offset:12"
                     :: "v"(dp), "v"(lds) : "memory");
        asm volatile("global_store_async_from_lds_b32 %0, %1, off offset:16"
                     :: "v"(dp), "v"(lds) : "memory");
        asm volatile("global_store_async_from_lds_b32 %0, %1, off offset:20"
                     :: "v"(dp), "v"(lds) : "memory");
        asm volatile("global_store_async_from_lds_b32 %0, %1, off offset:24"
                     :: "v"(dp), "v"(lds) : "memory");
        asm volatile("global_store_async_from_lds_b32 %0, %1, off offset:28"
                     :: "v"(dp), "v"(lds) : "memory");
        asm volatile("global_store_async_from_lds_b32 %0, %1, off offset:32"
                     :: "v"(dp), "v"(lds) : "memory");
        asm volatile("global_store_async_from_lds_b32 %0, %1, off offset:36"
                     :: "v"(dp), "v"(lds) : "memory");
        asm volatile("global_store_async_from_lds_b32 %0, %1, off offset:40"
                     :: "v"(dp), "v"(lds) : "memory");
        asm volatile("global_store_async_from_lds_b32 %0, %1, off offset:44"
                     :: "v"(dp), "v"(lds) : "memory");
        asm volatile("global_store_async_from_lds_b32 %0, %1, off offset:48"
                     :: "v"(dp), "v"(lds) : "memory");
        asm volatile("global_store_async_from_lds_b32 %0, %1, off offset:52"
                     :: "v"(dp), "v"(lds) : "memory");
        asm volatile("global_store_async_from_lds_b32 %0, %1, off offset:56"
                     :: "v"(dp), "v"(lds) : "memory");
        asm volatile("global_store_async_from_lds_b32 %0, %1, off offset:60"
                     :: "v"(dp), "v"(lds) : "memory");
        asm volatile("s_wait_asynccnt 0x0" ::: "memory");
    }
}

// ---------------------------------------------------------------------------
// Kernel 4: one wave (32 lanes) per sample, grid-stride over the batch.
// Each lane owns 8 of the 256 features: gather center row, diff, scatter
// -alpha*diff/(count+eps) with f32 atomics, and accumulate squared diffs into
// two per-lane registers. Final cross-lane reduction of the 64 per-wave
// partials is done with a single v_wmma_f32_16x16x4_f32 against an all-ones
// B matrix: D[m][n] = sum_k A[m][k], so the per-lane sum of the 8 result
// VGPRs yields the half-wave total in every lane of each 16-lane half.
// ---------------------------------------------------------------------------
__global__ void cl_scatter_loss_kernel(const float*  __restrict__ features,
                                       const int*    __restrict__ target,
                                       const float*  __restrict__ centers,
                                       const float*  __restrict__ counts,
                                       float*        __restrict__ out_centers,
                                       float*        __restrict__ out_loss) {
    const int lane   = threadIdx.x & 31;
    const int wave   = (blockIdx.x * blockDim.x + threadIdx.x) >> 5;
    const int nwaves = (gridDim.x * blockDim.x) >> 5;

    float acc0 = 0.0f, acc1 = 0.0f;

    for (int s = wave; s < BATCH; s += nwaves) {         // wave-uniform loop
        const int   t   = target[s];
        const float inv = ALPHA_C / (counts[t] + EPS_C);

        const float4* frow = (const float4*)(features + (size_t)s * FEAT_DIM);
        const float4* crow = (const float4*)(centers  + (size_t)t * FEAT_DIM);
        float*        orow = out_centers + (size_t)t * FEAT_DIM + (size_t)lane * 8;

        const float4 f0 = frow[lane * 2 + 0];
        const float4 f1 = frow[lane * 2 + 1];
        const float4 c0 = crow[lane * 2 + 0];
        const float4 c1 = crow[lane * 2 + 1];

        const float d0x = c0.x - f0.x, d0y = c0.y - f0.y, d0z = c0.z - f0.z, d0w = c0.w - f0.w;
        const float d1x = c1.x - f1.x, d1y = c1.y - f1.y, d1z = c1.z - f1.z, d1w = c1.w - f1.w;

        acc0 += d0x * d0x + d0y * d0y + d0z * d0z + d0w * d0w;
        acc1 += d1x * d1x + d1y * d1y + d1z * d1z + d1w * d1w;

        atomicAdd(orow + 0, -inv * d0x);
        atomicAdd(orow + 1, -inv * d0y);
        atomicAdd(orow + 2, -inv * d0z);
        atomicAdd(orow + 3, -inv * d0w);
        atomicAdd(orow + 4, -inv * d1x);
        atomicAdd(orow + 5, -inv * d1y);
        atomicAdd(orow + 6, -inv * d1z);
        atomicAdd(orow + 7, -inv * d1w);
    }

    // Cross-lane reduction via the matrix pipe (EXEC is all-ones here).
    v2f a;  a[0] = acc0;  a[1] = acc1;
    v2f b;  b[0] = 1.0f;  b[1] = 1.0f;
    v8f c = {};
    v8f d = __builtin_amdgcn_wmma_f32_16x16x4_f32(
        /*neg_a=*/false, a, /*neg_b=*/false, b,
        /*c_mod=*/(short)0, c, /*reuse_a=*/false, /*reuse_b=*/false);

    float half_sum = d[0] + d[1] + d[2] + d[3] + d[4] + d[5] + d[6] + d[7];

    // Lanes 0..15 hold sum over rows m=0..7; lanes 16..31 hold m=8..15.
    if (lane == 0 || lane == 16) {
        const float scale = 1.0f / ((float)BATCH * (float)FEAT_DIM);
        atomicAdd(out_loss, half_sum * scale);
    }
}

// ---------------------------------------------------------------------------
// Host-side launcher (graph-capture safe: kernels only, all on `stream`).
// ---------------------------------------------------------------------------
extern "C" void kernel_launch(void* const* d_in, const int* in_sizes, int n_in,
                              void* d_out, int out_size, void* d_ws, size_t ws_size,
                              hipStream_t stream) {
    const float* features = (const float*)d_in[0];   // [BATCH, FEAT_DIM] f32
    const int*   target   = (const int*)  d_in[1];   // [BATCH] int
    const float* centers  = (const float*)d_in[2];   // [NUM_CLASSES, FEAT_DIM] f32

    float* out_loss    = (float*)d_out;              // d_out[0] = center_loss
    float* out_centers = out_loss + 1;               // d_out[1..] = new_centers
    float* counts      = (float*)d_ws;               // NUM_CLASSES floats scratch

    // 1) zero counts + loss
    {
        int threads = 256;
        int blocks  = (NUM_CLASSES + threads - 1) / threads;
        cl_zero_kernel<<<blocks, threads, 0, stream>>>(counts, out_loss);
    }
    // 2) per-class counts
    {
        int threads = 256;
        int blocks  = (BATCH + threads - 1) / threads;
        cl_count_kernel<<<blocks, threads, 0, stream>>>(target, counts);
    }
    // 3) bulk centers -> new_centers via async LDS staging (64B per lane/window)
    {
        const int n16 = (NUM_CLASSES * FEAT_DIM) / 16;  // 1.6M 64-byte chunks
        cl_copy_kernel<<<2048, 256, 0, stream>>>(centers, out_centers, n16);
    }
    // 4) gather/diff/scatter + WMMA loss reduction
    {
        cl_scatter_loss_kernel<<<256, 256, 0, stream>>>(features, target, centers,
                                                        counts, out_centers, out_loss);
    }
}